// PointerWithRelations_35553739276501
// MI455X (gfx1250) — compile-verified
//
#include <hip/hip_runtime.h>
#include <hip/hip_bf16.h>
#include <math.h>

typedef __attribute__((ext_vector_type(2))) float v2f;
typedef __attribute__((ext_vector_type(8))) float v8f;

#define SQ   768   // sequence length
#define HD   256   // hidden size
#define NR   51    // relation kinds
#define RPAD 64    // padded relation width

// D = A(16x4 f32) * B(4x16 f32) + C(16x16 f32)
__device__ __forceinline__ v8f wmma_f32(v2f a, v2f b, v8f c) {
  // 8 args: (neg_a, A, neg_b, B, c_mod, C, reuse_a, reuse_b)
  return __builtin_amdgcn_wmma_f32_16x16x4_f32(
      false, a, false, b, (short)0, c, false, false);
}

// ---------------------------------------------------------------------------
// Kernel 1: Out = X @ W^T + bias    X:[SQ,HD], W:[HD,HD], Out:[SQ,HD]
// grid.x = (SQ/16 * HD/16) / 8 = 96 blocks of 8 waves, grid.y selects Q or K.
// ---------------------------------------------------------------------------
__global__ void proj_kernel(const float* __restrict__ query,
                            const float* __restrict__ key,
                            const float* __restrict__ Wq, const float* __restrict__ bq,
                            const float* __restrict__ Wk, const float* __restrict__ bk,
                            float* __restrict__ Qp, float* __restrict__ Kp) {
  const float *X, *W, *bias;
  float* Out;
  if (blockIdx.y == 0) { X = query; W = Wq; bias = bq; Out = Qp; }
  else                 { X = key;   W = Wk; bias = bk; Out = Kp; }

  const int wave = threadIdx.x >> 5;
  const int tile = blockIdx.x * 8 + wave;     // 0..767 = 48 mtiles * 16 ntiles
  const int m0 = (tile >> 4) * 16;
  const int n0 = (tile & 15) * 16;
  const int lane = threadIdx.x & 31;
  const int lr = lane & 15;                   // row-within-tile (A: M, B: N)
  const int kh = (lane >> 4) << 1;            // K sub-offset: 0 or 2

  const float* arow = X + (m0 + lr) * HD + kh;
  const float* brow = W + (n0 + lr) * HD + kh;  // B[k,n] = W[n,k] (X @ W^T)

  v8f acc = {};
  #pragma unroll 8
  for (int k0 = 0; k0 < HD; k0 += 4) {
    v2f a = *(const v2f*)(arow + k0);
    v2f b = *(const v2f*)(brow + k0);
    acc = wmma_f32(a, b, acc);
  }

  const float bn = bias[n0 + lr];
  const int mbase = (lane >> 4) << 3;         // 0 or 8
  #pragma unroll
  for (int v = 0; v < 8; ++v)
    Out[(m0 + mbase + v) * HD + n0 + lr] = acc[v] + bn;
}

// ---------------------------------------------------------------------------
// Kernel 2: QR[q, r] = Q[q,:] . rel_k_emb[r,:]   -> [SQ, RPAD] (cols >= NR junk)
// tiles: 48 m x 4 n = 192 -> 24 blocks of 8 waves
// ---------------------------------------------------------------------------
__global__ void qr_kernel(const float* __restrict__ Qp,
                          const float* __restrict__ relk,
                          float* __restrict__ QR) {
  const int wave = threadIdx.x >> 5;
  const int tile = blockIdx.x * 8 + wave;     // 0..191
  const int m0 = (tile >> 2) * 16;
  const int n0 = (tile & 3) * 16;
  const int lane = threadIdx.x & 31;
  const int lr = lane & 15;
  const int kh = (lane >> 4) << 1;
  const int n = n0 + lr;

  const float* arow = Qp + (m0 + lr) * HD + kh;
  const float* brow = relk + n * HD + kh;     // valid only when n < NR

  v8f acc = {};
  #pragma unroll 8
  for (int k0 = 0; k0 < HD; k0 += 4) {
    v2f a = *(const v2f*)(arow + k0);
    v2f b;
    if (n < NR) { b = *(const v2f*)(brow + k0); }
    else        { b[0] = 0.0f; b[1] = 0.0f; }
    acc = wmma_f32(a, b, acc);
  }

  const int mbase = (lane >> 4) << 3;
  #pragma unroll
  for (int v = 0; v < 8; ++v)
    QR[(m0 + mbase + v) * RPAD + n] = acc[v];
}

// ---------------------------------------------------------------------------
// Kernel 3: Sc = (Qp @ Kp^T + gather(QR, relation)) / sqrt(HD), masked.
// tiles: 48 x 48 = 2304 -> 288 blocks of 8 waves
// ---------------------------------------------------------------------------
__global__ void scores_kernel(const float* __restrict__ Qp,
                              const float* __restrict__ Kp,
                              const float* __restrict__ QR,
                              const int* __restrict__ relation,
                              const int* __restrict__ mask,
                              float* __restrict__ Sc) {
  const int wave = threadIdx.x >> 5;
  const int tile = blockIdx.x * 8 + wave;     // 0..2303
  const int m0 = (tile / 48) * 16;
  const int n0 = (tile % 48) * 16;
  const int lane = threadIdx.x & 31;
  const int lr = lane & 15;
  const int kh = (lane >> 4) << 1;

  const float* arow = Qp + (m0 + lr) * HD + kh;
  const float* brow = Kp + (n0 + lr) * HD + kh;  // B[k,n] = Kp[n,k]

  v8f acc = {};
  #pragma unroll 8
  for (int k0 = 0; k0 < HD; k0 += 4) {
    v2f a = *(const v2f*)(arow + k0);
    v2f b = *(const v2f*)(brow + k0);
    acc = wmma_f32(a, b, acc);
  }

  const int ncol = n0 + lr;
  const int mbase = (lane >> 4) << 3;
  const float scale = 0.0625f;                // 1/sqrt(256)
  #pragma unroll
  for (int v = 0; v < 8; ++v) {
    const int q = m0 + mbase + v;
    const int idx = q * SQ + ncol;
    const int r = relation[idx];              // 0..50
    float s = (acc[v] + QR[q * RPAD + r]) * scale;
    if (mask[idx] == 0) s = -1e9f;
    Sc[idx] = s;
  }
}

// ---------------------------------------------------------------------------
// Kernel 4: row softmax, one 256-thread block per row (3 cols per thread).
// ---------------------------------------------------------------------------
__global__ void softmax_kernel(const float* __restrict__ Sc,
                               float* __restrict__ out) {
  __shared__ float red[256];
  const int row = blockIdx.x;
  const int t = threadIdx.x;
  const float* srow = Sc + row * SQ;

  float vals[3];
  float lmax = -INFINITY;
  #pragma unroll
  for (int i = 0; i < 3; ++i) {
    vals[i] = srow[t + i * 256];
    lmax = fmaxf(lmax, vals[i]);
  }
  red[t] = lmax;
  __syncthreads();
  for (int s = 128; s > 0; s >>= 1) {
    if (t < s) red[t] = fmaxf(red[t], red[t + s]);
    __syncthreads();
  }
  const float m = red[0];
  __syncthreads();

  float lsum = 0.0f;
  #pragma unroll
  for (int i = 0; i < 3; ++i) {
    vals[i] = __expf(vals[i] - m);
    lsum += vals[i];
  }
  red[t] = lsum;
  __syncthreads();
  for (int s = 128; s > 0; s >>= 1) {
    if (t < s) red[t] += red[t + s];
    __syncthreads();
  }
  const float inv = 1.0f / red[0];
  #pragma unroll
  for (int i = 0; i < 3; ++i)
    out[row * SQ + t + i * 256] = vals[i] * inv;
}

// ---------------------------------------------------------------------------
// Launch. Input order (setup_inputs dict):
// 0 query, 1 key, 2 value, 3 relation, 4 mask,
// 5 Wq, 6 bq, 7 Wk, 8 bk, 9 Wv, 10 bv, 11 rel_k_emb, 12 rel_v_emb
// value/Wv/bv/rel_v_emb are dead code in the reference.
// ---------------------------------------------------------------------------
extern "C" void kernel_launch(void* const* d_in, const int* in_sizes, int n_in,
                              void* d_out, int out_size, void* d_ws, size_t ws_size,
                              hipStream_t stream) {
  const float* query    = (const float*)d_in[0];
  const float* key      = (const float*)d_in[1];
  const int*   relation = (const int*)d_in[3];
  const int*   mask     = (const int*)d_in[4];
  const float* Wq       = (const float*)d_in[5];
  const float* bq       = (const float*)d_in[6];
  const float* Wk       = (const float*)d_in[7];
  const float* bk       = (const float*)d_in[8];
  const float* relk     = (const float*)d_in[11];
  float* out = (float*)d_out;

  // Workspace layout (floats)
  float* ws = (float*)d_ws;
  float* Qp = ws;                         // SQ*HD   = 196608
  float* Kp = Qp + SQ * HD;               // SQ*HD   = 196608
  float* QR = Kp + SQ * HD;               // SQ*RPAD = 49152
  float* Sc = QR + SQ * RPAD;             // SQ*SQ   = 589824
  // total ~4.1 MB

  dim3 blk(256, 1, 1);

  // 1) Q and K projections: 768 tiles each, 8 waves/block -> 96 blocks, y = {Q,K}
  proj_kernel<<<dim3(96, 2, 1), blk, 0, stream>>>(query, key, Wq, bq, Wk, bk, Qp, Kp);

  // 2) QR = Q @ rel_k^T (padded to 64 cols): 192 tiles -> 24 blocks
  qr_kernel<<<dim3(24, 1, 1), blk, 0, stream>>>(Qp, relk, QR);

  // 3) scores = Q@K^T + gather(QR, relation), scaled + masked: 2304 tiles -> 288 blocks
  scores_kernel<<<dim3(288, 1, 1), blk, 0, stream>>>(Qp, Kp, QR, relation, mask, Sc);

  // 4) row softmax -> out
  softmax_kernel<<<dim3(SQ, 1, 1), blk, 0, stream>>>(Sc, out);
}